// FeatureLoss_43413529428715
// MI455X (gfx1250) — compile-verified
//
#include <hip/hip_runtime.h>
#include <hip/hip_bf16.h>

// ---------------- problem constants (fixed by the reference) ----------------
constexpr int B_    = 2;
constexpr int C_    = 64;
constexpr int H_    = 512;
constexpr int W_    = 512;
constexpr long HW_  = (long)H_ * W_;
constexpr int N_    = 4096;
constexpr int K_    = 5;                 // negatives per match
constexpr int RPM   = K_ + 1;            // rows per match (1 pos + 5 neg)
constexpr float MAXDIST    = 724.0773439f;        // sqrt(512^2+512^2)
constexpr float THR_NOMATCH = 16.0f;              // 0.5^2 * 64
constexpr float SIZE_AVG    = 1.0f / 24576.0f;    // 1/((K+1)*N)

// block geometry: 16 matches -> 96 rows -> 6 waves of 16 rows
constexpr int MATCHES_PER_BLOCK = 16;
constexpr int ROWS   = MATCHES_PER_BLOCK * RPM;   // 96
constexpr int RSTR   = 68;                        // padded LDS row stride (floats)
constexpr int THREADS = 192;                      // 6 wave32

typedef float v2f __attribute__((ext_vector_type(2)));
typedef float v8f __attribute__((ext_vector_type(8)));

__global__ void __launch_bounds__(THREADS)
FeatureLoss_43413529428715_main(const float* __restrict__ inputs_ref,
                                const float* __restrict__ inputs_other,
                                const float* __restrict__ weights,
                                const int*   __restrict__ inds_ref,
                                const int*   __restrict__ inds_other,
                                const int*   __restrict__ rand_inds,
                                float*       __restrict__ partials) {
    __shared__ __align__(16) float Dlds[ROWS * RSTR];  // 96*68*4 = 26112 B
    __shared__ float waveSum[THREADS / 32];

    const int t  = threadIdx.x;
    const int b  = blockIdx.x / (N_ / MATCHES_PER_BLOCK);
    const int m0 = (blockIdx.x % (N_ / MATCHES_PER_BLOCK)) * MATCHES_PER_BLOCK;

    // ---------------- phase 1: gather difference rows into LDS ----------------
    // D[row][c] = fr[b, c, pix_ref(i)] - fo[b, c, pix_other(j(row))]
    for (int e = t; e < ROWS * C_; e += THREADS) {
        const int row = e >> 6;          // /64
        const int c   = e & 63;
        const int il  = row / RPM;
        const int k   = row - il * RPM;  // 0 = positive partner, 1..5 negatives
        const int i   = m0 + il;

        const int xr = inds_ref[(b * 2 + 0) * N_ + i];
        const int yr = inds_ref[(b * 2 + 1) * N_ + i];
        const long pr = (long)H_ * yr + xr;

        const int j  = (k == 0) ? i : rand_inds[((long)b * N_ + i) * K_ + (k - 1)];
        const int xo = inds_other[(b * 2 + 0) * N_ + j];
        const int yo = inds_other[(b * 2 + 1) * N_ + j];
        const long po = (long)H_ * yo + xo;

        const float fr = inputs_ref [((long)(b * C_ + c)) * HW_ + pr];
        const float fo = inputs_other[((long)(b * C_ + c)) * HW_ + po];
        Dlds[row * RSTR + c] = fr - fo;
    }
    __syncthreads();

    // ---------------- phase 2: squared L2 per row via WMMA self-product -------
    // A (16x4 f32): lanes 0-15 -> M=lane, V0/V1 = K0/K1 ; lanes 16-31 -> M=lane-16, K2/K3.
    // B (4x16 f32) has the identical register layout when B = A^T, so passing the
    // same registers for A and B yields D = A*A^T; diag(D) = row squared norms.
    const int wave = t >> 5;
    const int lane = t & 31;
    const int base = wave * 16;
    const int rowA = base + (lane & 15);
    const int c0   = (lane < 16) ? 0 : 2;

    const float2* rp2 = reinterpret_cast<const float2*>(&Dlds[rowA * RSTR + c0]);

    v8f acc = {};
    #pragma unroll
    for (int jj = 0; jj < 16; ++jj) {          // 16 chunks of K=4 cover C=64
        const float2 v = rp2[2 * jj];          // ds_load_b64
        v2f a;
        a.x = v.x;
        a.y = v.y;
        acc = __builtin_amdgcn_wmma_f32_16x16x4_f32(
            /*neg_a=*/false, a, /*neg_b=*/false, a,
            /*c_mod=*/(short)0, acc, /*reuse_a=*/false, /*reuse_b=*/false);
    }

    // diag(D) lives in: lanes 0-7 -> acc[lane] (rows base+0..7)
    //                   lanes 24-31 -> acc[lane-24] (rows base+8..15)
    float contrib = 0.0f;
    const bool active = (lane < 8) | (lane >= 24);
    if (active) {
        const int rl  = (lane < 8) ? lane : (lane - 16);  // row offset in tile
        const int idx = (lane < 8) ? lane : (lane - 24);  // acc element
        float s = acc[0];
        #pragma unroll
        for (int e2 = 1; e2 < 8; ++e2) s = (idx == e2) ? acc[e2] : s;

        const int row = base + rl;
        const int il  = row / RPM;
        const int k   = row - il * RPM;
        const int i   = m0 + il;

        if (k == 0) {
            contrib = weights[(long)b * N_ + i] * fmaxf(s, 0.0f);
        } else {
            const int j  = rand_inds[((long)b * N_ + i) * K_ + (k - 1)];
            const float xi = (float)inds_ref[(b * 2 + 0) * N_ + i];
            const float yi = (float)inds_ref[(b * 2 + 1) * N_ + i];
            const float xj = (float)inds_ref[(b * 2 + 0) * N_ + j];
            const float yj = (float)inds_ref[(b * 2 + 1) * N_ + j];
            const float dx = xi - xj, dy = yi - yj;
            const float dist = sqrtf(dx * dx + dy * dy);
            // w_mm = -NON_MATCH_WEIGHT * d / max_dist / K == -d / max_dist
            contrib = -(dist / MAXDIST) * fminf(s, THR_NOMATCH);
        }
    }

    // fixed-tree wave32 reduction (deterministic)
    #pragma unroll
    for (int off = 16; off > 0; off >>= 1)
        contrib += __shfl_xor(contrib, off, 32);
    if (lane == 0) waveSum[wave] = contrib;
    __syncthreads();

    if (t == 0) {
        float ssum = 0.0f;
        #pragma unroll
        for (int wv = 0; wv < THREADS / 32; ++wv) ssum += waveSum[wv];
        partials[blockIdx.x] = ssum * SIZE_AVG;
    }
}

// deterministic final reduction of 512 block partials
__global__ void __launch_bounds__(256)
FeatureLoss_43413529428715_reduce(const float* __restrict__ partials,
                                  float* __restrict__ out) {
    __shared__ float sm[256];
    const int t = threadIdx.x;
    sm[t] = partials[t] + partials[t + 256];
    __syncthreads();
    #pragma unroll
    for (int s = 128; s > 0; s >>= 1) {
        if (t < s) sm[t] += sm[t + s];
        __syncthreads();
    }
    if (t == 0) out[0] = sm[0];
}

extern "C" void kernel_launch(void* const* d_in, const int* in_sizes, int n_in,
                              void* d_out, int out_size, void* d_ws, size_t ws_size,
                              hipStream_t stream) {
    const float* inputs_ref   = (const float*)d_in[0];
    const float* inputs_other = (const float*)d_in[1];
    const float* weights      = (const float*)d_in[2];
    const int*   inds_ref     = (const int*)d_in[3];
    const int*   inds_other   = (const int*)d_in[4];
    const int*   rand_inds    = (const int*)d_in[5];
    float* partials = (float*)d_ws;         // 512 floats
    float* out      = (float*)d_out;

    const int numBlocks = (B_ * N_) / MATCHES_PER_BLOCK;  // 512
    FeatureLoss_43413529428715_main<<<dim3(numBlocks), dim3(THREADS), 0, stream>>>(
        inputs_ref, inputs_other, weights, inds_ref, inds_other, rand_inds, partials);
    FeatureLoss_43413529428715_reduce<<<dim3(1), dim3(256), 0, stream>>>(partials, out);
}